// CPQuadRankLayer_82411832475666
// MI455X (gfx1250) — compile-verified
//
#include <hip/hip_runtime.h>

// CPQuadRankLayer for MI455X (gfx1250, wave32).
// B=64, N=1024, R=64, D=128, f32. Memory-bound (~320MiB @23.3TB/s ~= 14us floor).
// One workgroup per node n (256 thr = 8 waves):
//   Phase A (c=0..3): projected_c = X_c(64x128) * F_c^T -> 64x64 via v_wmma_f32_16x16x4_f32,
//                     row-RMS (ds_add_f32 atomics), running elementwise product in regs.
//   Phase B: out = merged(64x64) * Fout(64x128) + mean_c(x), residual seeded into C.
// Global->LDS staging uses double-buffered GLOBAL_LOAD_ASYNC_TO_LDS_B128 (ASYNCcnt),
// so the prefetch of iteration c+1 (and factor_out at c=3) overlaps WMMA compute of c.

typedef __attribute__((ext_vector_type(2))) float v2f;
typedef __attribute__((ext_vector_type(8))) float v8f;

#define XS 132   // padded LDS row stride for 128-wide buffers (conflict-free WMMA frag loads)
#define MS 68    // padded LDS row stride for the 64-wide merged buffer

// Issue this thread's share of one 64x128 f32 tile: global (rowStride elts) -> LDS (stride XS),
// as 8 async 16B copies. No VGPR round trip; tracked by ASYNCcnt.
__device__ __forceinline__ void issue_tile_async(const float* gslice, uint32_t rowStrideElts,
                                                 uint32_t ldsBase, int t) {
  unsigned long long base = (unsigned long long)(uintptr_t)gslice;
#pragma unroll
  for (int j = 0; j < 8; ++j) {
    int m = j * 256 + t;            // 16B-chunk index 0..2047 (wave covers one contiguous row)
    int row = m >> 5;               // 32 chunks per 128-float row
    int col = (m & 31) << 2;
    uint32_t goff = (uint32_t)(row * rowStrideElts + (uint32_t)col) * 4u;
    uint32_t ldst = ldsBase + (uint32_t)(row * XS + col) * 4u;
    asm volatile("global_load_async_to_lds_b128 %0, %1, %2"
                 :: "v"(ldst), "v"(goff), "s"(base) : "memory");
  }
}

__global__ __launch_bounds__(256) void cpquad_kernel(
    const float* __restrict__ x,        // [64,1024,4,128]
    const float* __restrict__ factors,  // [4,1024,64,128]
    const float* __restrict__ fout,     // [1024,64,128]
    const float* __restrict__ gain,     // [1024,1]
    float* __restrict__ out) {          // [64,1024,128]
  constexpr int N = 1024;
  const int n = blockIdx.x;
  const int t = threadIdx.x;
  const int wave = t >> 5;
  const int lane = t & 31;
  const int l16 = lane & 15;
  const int half = lane >> 4;           // 0/1
  const int rg  = wave >> 1;            // b row-group (0..3)
  const int cg0 = (wave & 1) * 2;       // first r col-group (0 or 2)

  __shared__ float sX[2][64 * XS];   // x slice (double buffered); sX[0] -> residual in phase B
  __shared__ float sF[2][64 * XS];   // factors (double buffered); sF[0] -> factor_out in phase B
  __shared__ float sM[64 * MS];      // merged [b][r]
  __shared__ float sRow[64];         // sum of squares per b row

  const uint32_t ldsX[2] = { (uint32_t)(uintptr_t)&sX[0][0], (uint32_t)(uintptr_t)&sX[1][0] };
  const uint32_t ldsF[2] = { (uint32_t)(uintptr_t)&sF[0][0], (uint32_t)(uintptr_t)&sF[1][0] };
  const uint32_t xStride = (uint32_t)N * 4u * 128u;   // elements between b rows

  float prod0[8], prod1[8];
#pragma unroll
  for (int v = 0; v < 8; ++v) { prod0[v] = 1.0f; prod1[v] = 1.0f; }
  float racc[32];
#pragma unroll
  for (int k = 0; k < 32; ++k) racc[k] = 0.0f;

  const int resB = t >> 2;              // this thread's residual row b
  const int resO = (t & 3) * 32;        // this thread's residual col base

  // Prologue: prefetch c=0 into buffer 0.
  issue_tile_async(x + ((size_t)n * 4 + 0) * 128, xStride, ldsX[0], t);
  issue_tile_async(factors + (((size_t)0 * N + n) * 64) * 128, 128u, ldsF[0], t);

  // ---------------- Phase A ------------------------------------------------------
  for (int c = 0; c < 4; ++c) {
    const int cur = c & 1, nxt = cur ^ 1;
    __syncthreads();  // all waves done reading buf[nxt] (used at c-1); sRow reads done
    if (c < 3) {      // prefetch next c into the other buffer
      issue_tile_async(x + ((size_t)n * 4 + (c + 1)) * 128, xStride, ldsX[nxt], t);
      issue_tile_async(factors + (((size_t)(c + 1) * N + n) * 64) * 128, 128u, ldsF[nxt], t);
    } else {          // prefetch factor_out for phase B into sF[0]
      issue_tile_async(fout + (size_t)n * 64 * 128, 128u, ldsF[nxt], t);
    }
    if (t < 64) sRow[t] = 0.0f;
    // Async loads complete in order: waiting <= (just-issued count) means cur has landed.
    if (c < 3) { asm volatile("s_wait_asynccnt 0x10" ::: "memory"); }
    else       { asm volatile("s_wait_asynccnt 0x8"  ::: "memory"); }
    __syncthreads();  // every wave's chunks of cur landed; sRow reinit visible

    const float* sA = sX[cur];
    const float* sBf = sF[cur];

    // residual accumulation (mean over c later)
#pragma unroll
    for (int k = 0; k < 32; k += 4) {
      float4 v = *(const float4*)&sA[resB * XS + resO + k];
      racc[k] += v.x; racc[k+1] += v.y; racc[k+2] += v.z; racc[k+3] += v.w;
    }

    // GEMM: two 16x16 tiles per wave, K=128 in 32 steps of k=4
    v8f acc0 = {}, acc1 = {};
    const float* aRow  = &sA [(rg * 16 + l16) * XS + half * 2];
    const float* bRow0 = &sBf[((cg0 + 0) * 16 + l16) * XS + half * 2];
    const float* bRow1 = &sBf[((cg0 + 1) * 16 + l16) * XS + half * 2];
#pragma unroll 8
    for (int s = 0; s < 32; ++s) {
      v2f a  = *(const v2f*)(aRow  + s * 4);
      v2f b0 = *(const v2f*)(bRow0 + s * 4);
      v2f b1 = *(const v2f*)(bRow1 + s * 4);
      acc0 = __builtin_amdgcn_wmma_f32_16x16x4_f32(false, a, false, b0, (short)0, acc0, false, false);
      acc1 = __builtin_amdgcn_wmma_f32_16x16x4_f32(false, a, false, b1, (short)0, acc1, false, false);
    }

    // row-wise sum of squares over rank dim (lds atomics, one bank per row)
#pragma unroll
    for (int v = 0; v < 8; ++v) {
      float ss = acc0[v] * acc0[v] + acc1[v] * acc1[v];
      atomicAdd(&sRow[rg * 16 + v + half * 8], ss);
    }
    __syncthreads();
    if (t < 64) sRow[t] = rsqrtf(sRow[t] * (1.0f / 64.0f) + 1e-6f);
    __syncthreads();
#pragma unroll
    for (int v = 0; v < 8; ++v) {
      float ri = sRow[rg * 16 + v + half * 8];
      prod0[v] *= acc0[v] * ri;
      prod1[v] *= acc1[v] * ri;
    }
  }

  // ---------------- Phase B: out = merged * Fout + mean_c(x) ----------------------
  asm volatile("s_wait_asynccnt 0x0" ::: "memory");  // this wave's factor_out chunks landed
  __syncthreads();  // all waves: phase-A reads done, factor_out fully in sF[0]
  const float g = gain[n];
#pragma unroll
  for (int v = 0; v < 8; ++v) {
    int row = rg * 16 + v + half * 8;
    sM[row * MS + (cg0 + 0) * 16 + l16] = prod0[v] * g;
    sM[row * MS + (cg0 + 1) * 16 + l16] = prod1[v] * g;
  }
  float* sR = sX[0];                 // residual buffer (free now)
#pragma unroll
  for (int k = 0; k < 32; ++k) sR[resB * XS + resO + k] = racc[k] * 0.25f;
  const float* sBo = sF[0];          // factor_out
  __syncthreads();

  // 4 output tiles per wave: rows rg*16.., cols ((wave&1)*4+q)*16..
  v8f o0 = {}, o1 = {}, o2 = {}, o3 = {};
  v8f* oacc[4] = {&o0, &o1, &o2, &o3};
#pragma unroll
  for (int q = 0; q < 4; ++q) {
    int col = ((wave & 1) * 4 + q) * 16 + l16;
#pragma unroll
    for (int v = 0; v < 8; ++v)
      (*oacc[q])[v] = sR[(rg * 16 + v + half * 8) * XS + col];   // seed C with residual
  }
  const float* aRow2 = &sM[(rg * 16 + l16) * MS + half * 2];
#pragma unroll 4
  for (int s = 0; s < 16; ++s) {
    v2f a = *(const v2f*)(aRow2 + s * 4);
#pragma unroll
    for (int q = 0; q < 4; ++q) {
      int col = ((wave & 1) * 4 + q) * 16 + l16;
      v2f bb;
      bb.x = sBo[(s * 4 + half * 2 + 0) * XS + col];
      bb.y = sBo[(s * 4 + half * 2 + 1) * XS + col];
      *oacc[q] = __builtin_amdgcn_wmma_f32_16x16x4_f32(false, a, false, bb, (short)0, *oacc[q], false, false);
    }
  }
#pragma unroll
  for (int q = 0; q < 4; ++q) {
    int col = ((wave & 1) * 4 + q) * 16 + l16;
#pragma unroll
    for (int v = 0; v < 8; ++v) {
      int b = rg * 16 + v + half * 8;
      out[((size_t)b * N + n) * 128 + col] = (*oacc[q])[v];
    }
  }
}

extern "C" void kernel_launch(void* const* d_in, const int* in_sizes, int n_in,
                              void* d_out, int out_size, void* d_ws, size_t ws_size,
                              hipStream_t stream) {
  const float* x       = (const float*)d_in[0];
  const float* factors = (const float*)d_in[1];
  const float* fout    = (const float*)d_in[2];
  const float* gain    = (const float*)d_in[3];
  float* out = (float*)d_out;
  (void)in_sizes; (void)n_in; (void)out_size; (void)d_ws; (void)ws_size;
  hipLaunchKernelGGL(cpquad_kernel, dim3(1024), dim3(256), 0, stream,
                     x, factors, fout, gain, out);
}